// GraphPolicyNetwork_31404800868906
// MI455X (gfx1250) — compile-verified
//
#include <hip/hip_runtime.h>

// GraphSAGE 2-layer (mean) for MI455X / gfx1250, wave32 + f32 WMMA.
//
// d_in order: x, W_self1, W_neigh1, b1, W_self2, W_neigh2, b2, src, dst
// d_out: logits [N, 64] float32
// d_ws layout: deg[N] f32 | agg[N*128] f32 | h[N*128] f32  (~51.4 MB)

typedef float v2f __attribute__((ext_vector_type(2)));
typedef float v8f __attribute__((ext_vector_type(8)));

static __device__ __forceinline__ v8f wmma_f32_16x16x4(v2f a, v2f b, v8f c) {
  // 8 args: (neg_a, A, neg_b, B, c_mod, C, reuse_a, reuse_b)
  return __builtin_amdgcn_wmma_f32_16x16x4_f32(
      /*neg_a=*/false, a, /*neg_b=*/false, b,
      /*c_mod=*/(short)0, c, /*reuse_a=*/false, /*reuse_b=*/false);
}

__global__ void zero_f32_kernel(float* __restrict__ p, size_t n) {
  size_t i = (size_t)blockIdx.x * blockDim.x + threadIdx.x;
  size_t stride = (size_t)gridDim.x * blockDim.x;
  for (; i < n; i += stride) p[i] = 0.0f;
}

__global__ void degree_kernel(const int* __restrict__ dst,
                              float* __restrict__ deg, int nE) {
  int e = blockIdx.x * blockDim.x + threadIdx.x;
  if (e < nE) atomicAdd(&deg[dst[e]], 1.0f);
}

// One wave32 per edge; lane l handles features [4l, 4l+4). 128 feats total.
__global__ void scatter_add_kernel(const float* __restrict__ x,
                                   const int* __restrict__ src,
                                   const int* __restrict__ dst,
                                   float* __restrict__ agg, int nE) {
  int gid  = blockIdx.x * blockDim.x + threadIdx.x;
  int e    = gid >> 5;
  int lane = gid & 31;
  if (e >= nE) return;
  int s = src[e];
  int d = dst[e];
  const float4 v = *(const float4*)(x + (size_t)s * 128 + lane * 4);
  float* o = agg + (size_t)d * 128 + lane * 4;
  atomicAdd(o + 0, v.x);
  atomicAdd(o + 1, v.y);
  atomicAdd(o + 2, v.z);
  atomicAdd(o + 3, v.w);
}

// One wave per 16x16 output tile:
//   out = X @ Wself + (AGG * 1/max(deg,1)) @ Wneigh + bias, optional ReLU.
// K = 128 fixed. nRows % 16 == 0 and nCols % 16 == 0 are required.
__global__ __launch_bounds__(256)
void sage_wmma_gemm_kernel(const float* __restrict__ X,
                           const float* __restrict__ AGG,
                           const float* __restrict__ DEG,
                           const float* __restrict__ Wself,
                           const float* __restrict__ Wneigh,
                           const float* __restrict__ bias,
                           float* __restrict__ out,
                           int nRows, int nCols, int doRelu) {
  const int wavesPerBlock = blockDim.x >> 5;
  const int wave = threadIdx.x >> 5;
  const int lane = threadIdx.x & 31;
  const int nTilesN = nCols >> 4;
  const int nTiles  = (nRows >> 4) * nTilesN;
  const int tile = blockIdx.x * wavesPerBlock + wave;
  if (tile >= nTiles) return;  // wave-uniform: whole wave exits, EXEC stays all-1s

  const int mt   = tile / nTilesN;
  const int nt   = tile - mt * nTilesN;
  const int half = lane >> 4;        // 0: lanes 0-15, 1: lanes 16-31
  const int l15  = lane & 15;
  const int aRow = mt * 16 + l15;    // A-matrix row for this lane
  const int bCol = nt * 16 + l15;    // B/C/D column for this lane
  const int kb   = half * 2;         // K sub-offset per ISA A/B layout

  v8f c = {0.f, 0.f, 0.f, 0.f, 0.f, 0.f, 0.f, 0.f};

  // ---- pass 1: X @ Wself ----
  const float* xr = X + (size_t)aRow * 128 + kb;           // contiguous K pair
  const float* wl = Wself + (size_t)kb * nCols + bCol;     // coalesced over lanes
#pragma unroll 8
  for (int k = 0; k < 128; k += 4) {
    v2f a = *(const v2f*)(xr + k);
    v2f b;
    b.x = wl[(size_t)k * nCols];
    b.y = wl[(size_t)(k + 1) * nCols];
    c = wmma_f32_16x16x4(a, b, c);
  }

  // ---- pass 2: (AGG / max(deg,1)) @ Wneigh, mean folded into A-load ----
  const float inv = 1.0f / fmaxf(DEG[aRow], 1.0f);
  const float* gr = AGG + (size_t)aRow * 128 + kb;
  const float* wn = Wneigh + (size_t)kb * nCols + bCol;
#pragma unroll 8
  for (int k = 0; k < 128; k += 4) {
    v2f a = *(const v2f*)(gr + k);
    a.x *= inv;
    a.y *= inv;
    v2f b;
    b.x = wn[(size_t)k * nCols];
    b.y = wn[(size_t)(k + 1) * nCols];
    c = wmma_f32_16x16x4(a, b, c);
  }

  // ---- epilogue: bias + optional ReLU, store 16x16 tile ----
  const float bv = bias[bCol];
  float* orow = out + (size_t)(mt * 16 + half * 8) * nCols + bCol;
#pragma unroll
  for (int i = 0; i < 8; ++i) {
    float v = c[i] + bv;
    if (doRelu) v = fmaxf(v, 0.0f);
    orow[(size_t)i * nCols] = v;
  }
}

extern "C" void kernel_launch(void* const* d_in, const int* in_sizes, int n_in,
                              void* d_out, int out_size, void* d_ws, size_t ws_size,
                              hipStream_t stream) {
  const float* x   = (const float*)d_in[0];
  const float* Ws1 = (const float*)d_in[1];
  const float* Wn1 = (const float*)d_in[2];
  const float* b1  = (const float*)d_in[3];
  const float* Ws2 = (const float*)d_in[4];
  const float* Wn2 = (const float*)d_in[5];
  const float* b2  = (const float*)d_in[6];
  const int*   src = (const int*)d_in[7];
  const int*   dst = (const int*)d_in[8];
  float* out = (float*)d_out;

  const int F = in_sizes[3];               // 128 (hidden == feat)
  const int C = in_sizes[6];               // 64
  const int N = in_sizes[0] / F;           // 50000 (divisible by 16)
  const int E = in_sizes[7];               // 800000

  char* ws = (char*)d_ws;
  float* deg = (float*)ws;                                   // N floats
  float* agg = (float*)(ws + (size_t)N * 4);                 // N*F floats
  float* h   = (float*)(ws + (size_t)N * 4 + (size_t)N * F * 4);  // N*F floats

  const int ZB = 2048, TB = 256;
  const int edgeWaveBlocks = (E * 32 + TB - 1) / TB;
  const int edgeBlocks     = (E + TB - 1) / TB;

  // ---------------- layer 1 ----------------
  // zero deg + agg (contiguous in ws)
  zero_f32_kernel<<<ZB, TB, 0, stream>>>(deg, (size_t)N + (size_t)N * F);
  degree_kernel<<<edgeBlocks, TB, 0, stream>>>(dst, deg, E);
  scatter_add_kernel<<<edgeWaveBlocks, TB, 0, stream>>>(x, src, dst, agg, E);

  {
    const int nTiles = (N / 16) * (F / 16);
    const int blocks = (nTiles + (TB / 32) - 1) / (TB / 32);
    sage_wmma_gemm_kernel<<<blocks, TB, 0, stream>>>(
        x, agg, deg, Ws1, Wn1, b1, h, N, F, /*doRelu=*/1);
  }

  // ---------------- layer 2 ----------------
  zero_f32_kernel<<<ZB, TB, 0, stream>>>(agg, (size_t)N * F);
  scatter_add_kernel<<<edgeWaveBlocks, TB, 0, stream>>>(h, src, dst, agg, E);

  {
    const int nTiles = (N / 16) * (C / 16);
    const int blocks = (nTiles + (TB / 32) - 1) / (TB / 32);
    sage_wmma_gemm_kernel<<<blocks, TB, 0, stream>>>(
        h, agg, deg, Ws2, Wn2, b2, out, N, C, /*doRelu=*/0);
  }
}